// GNN_atom_bond_43387759624513
// MI455X (gfx1250) — compile-verified
//
#include <hip/hip_runtime.h>
#include <hip/hip_bf16.h>

typedef float v2f __attribute__((ext_vector_type(2)));
typedef float v8f __attribute__((ext_vector_type(8)));
typedef int   v4i __attribute__((ext_vector_type(4)));

#define GLOBAL_AS __attribute__((address_space(1)))
#define LDS_AS    __attribute__((address_space(3)))

#define D_DIM 256
#define B_DIM 16
#define N_DIM 256
#define L_DIM 1024
#define ATT_STRIDE 1025   // softmax rows padded: stride%64 = 1 -> conflict-free row readers
#define A_STRIDE   260    // proj A-tile rows padded: stride%64 = 4 -> conflict-free fragment reads

#if __has_builtin(__builtin_amdgcn_global_load_async_to_lds_b128) && \
    __has_builtin(__builtin_amdgcn_s_wait_asynccnt)
#define USE_ASYNC_LDS 1
#else
#define USE_ASYNC_LDS 0
#endif

// ---------------------------------------------------------------------------
// Kernel 0: fold projections through align_w.
//   v[d] = sum_e W_mol[e,d] * w1[e],  u[d] = sum_e W_nb[e,d] * w2[e]
//   c12[0] = b_mol . w1, c12[1] = b_nb . w2
// ---------------------------------------------------------------------------
__global__ __launch_bounds__(256) void precompute_kernel(
    const float* __restrict__ Wmol, const float* __restrict__ Wnb,
    const float* __restrict__ alignw,
    const float* __restrict__ bmol, const float* __restrict__ bnb,
    float* __restrict__ v, float* __restrict__ u, float* __restrict__ c12)
{
    const int d = threadIdx.x;
    float sv = 0.f, su = 0.f;
    for (int e = 0; e < D_DIM; ++e) {
        sv += Wmol[e * D_DIM + d] * alignw[e];
        su += Wnb [e * D_DIM + d] * alignw[D_DIM + e];
    }
    v[d] = sv;
    u[d] = su;
    if (d == 0) { float s = 0.f; for (int e = 0; e < D_DIM; ++e) s += bmol[e] * alignw[e];          c12[0] = s; }
    if (d == 1) { float s = 0.f; for (int e = 0; e < D_DIM; ++e) s += bnb[e]  * alignw[D_DIM + e];  c12[1] = s; }
}

// ---------------------------------------------------------------------------
// Kernel 1: a = atom_feature @ W_nb^T + b_nb   (16384 x 256 x 256, f32 WMMA)
// One block (16 waves, 512 thr) per 16-row m-tile; wave w owns column tile w.
// A tile (16x256) staged once in LDS via async-to-LDS; B streamed from L2.
// ---------------------------------------------------------------------------
__global__ __launch_bounds__(512) void proj_a_kernel(
    const float* __restrict__ atom, const float* __restrict__ Wnb,
    const float* __restrict__ bnb, float* __restrict__ a_out)
{
    __shared__ float lds_a[16 * A_STRIDE];     // 16.6 KB

    const int mtile = blockIdx.x;              // 0..1023
    const int tid   = threadIdx.x;
    const int lane  = threadIdx.x & 31;
    const int wave  = threadIdx.x >> 5;        // = ntile, 0..15

    // ---- stage A tile: 16 rows x 256 floats = 1024 x 16B chunks ----
#if USE_ASYNC_LDS
    for (int c = tid; c < 1024; c += 512) {
        const int m  = c >> 6;
        const int kc = (c & 63) << 2;
        const float* gp = atom + ((size_t)(mtile * 16 + m)) * D_DIM + kc;
        float*       lp = &lds_a[m * A_STRIDE + kc];
        __builtin_amdgcn_global_load_async_to_lds_b128(
            (GLOBAL_AS v4i*)gp, (LDS_AS v4i*)lp, 0, 0);
    }
    __builtin_amdgcn_s_wait_asynccnt(0);
#else
    for (int c = tid; c < 1024; c += 512) {
        const int m  = c >> 6;
        const int kc = (c & 63) << 2;
        *(float4*)&lds_a[m * A_STRIDE + kc] =
            *(const float4*)(atom + ((size_t)(mtile * 16 + m)) * D_DIM + kc);
    }
#endif
    __syncthreads();

    // ---- WMMA: 16x16 tile per wave, K = 256 ----
    const int col  = wave * 16 + (lane & 15);      // output column (e)
    const int koff = (lane >> 4) << 1;             // half-wave K offset (0 or 2)
    const float* __restrict__ wcol = Wnb + (size_t)col * D_DIM;   // B[k,n] = W_nb[n,k]
    const float* __restrict__ arow = &lds_a[(lane & 15) * A_STRIDE];

    v8f c = {};
#pragma unroll 4
    for (int k = 0; k < D_DIM; k += 4) {
        v2f av, bv;
        av.x = arow[k + koff];
        av.y = arow[k + koff + 1];
        bv.x = wcol[k + koff];
        bv.y = wcol[k + koff + 1];
        c = __builtin_amdgcn_wmma_f32_16x16x4_f32(false, av, false, bv,
                                                  (short)0, c, false, false);
    }

    const float bias  = bnb[col];
    const int   mbase = mtile * 16 + ((lane >> 4) << 3);  // M=r (lanes 0-15) / r+8 (16-31)
#pragma unroll
    for (int r = 0; r < 8; ++r)
        a_out[(size_t)(mbase + r) * D_DIM + col] = c[r] + bias;
}

// ---------------------------------------------------------------------------
// Kernel 2: xw1[row] = mol[row] . v + c1 ; aw2[row] = atom[row] . u + c2
// ---------------------------------------------------------------------------
__global__ __launch_bounds__(256) void dots_kernel(
    const float* __restrict__ mol, const float* __restrict__ atom,
    const float* __restrict__ v, const float* __restrict__ u,
    const float* __restrict__ c12,
    float* __restrict__ xw1, float* __restrict__ aw2)
{
    const int idx = blockIdx.x * blockDim.x + threadIdx.x;   // 0..20479
    if (idx < B_DIM * N_DIM) {
        const float4* r4 = (const float4*)(mol + (size_t)idx * D_DIM);
        const float4* v4 = (const float4*)v;
        float s = 0.f;
#pragma unroll 8
        for (int i = 0; i < D_DIM / 4; ++i) {
            float4 x = r4[i], y = v4[i];
            s += x.x * y.x + x.y * y.y + x.z * y.z + x.w * y.w;
        }
        xw1[idx] = s + c12[0];
    } else {
        const int j = idx - B_DIM * N_DIM;
        const float4* r4 = (const float4*)(atom + (size_t)j * D_DIM);
        const float4* u4 = (const float4*)u;
        float s = 0.f;
#pragma unroll 8
        for (int i = 0; i < D_DIM / 4; ++i) {
            float4 x = r4[i], y = u4[i];
            s += x.x * y.x + x.y * y.y + x.z * y.z + x.w * y.w;
        }
        aw2[j] = s + c12[1];
    }
}

// ---------------------------------------------------------------------------
// Kernel 3: fused score -> leaky_relu -> softmax -> mask -> ctx WMMA -> LN
// One block per (b, 16-row n-tile). 512 threads = 16 waves.
// ---------------------------------------------------------------------------
__global__ __launch_bounds__(512) void attn_kernel(
    const float* __restrict__ a, const float* __restrict__ aw2,
    const float* __restrict__ xw1,
    const float* __restrict__ attend_mask, const float* __restrict__ softmax_mask,
    const float* __restrict__ alignb,
    const float* __restrict__ gamma, const float* __restrict__ beta,
    float* __restrict__ out)
{
    extern __shared__ float lds[];                 // 16 * ATT_STRIDE floats
    const int b     = blockIdx.x >> 4;
    const int ntile = blockIdx.x & 15;
    const int wave  = threadIdx.x >> 5;
    const int lane  = threadIdx.x & 31;
    const float ab  = alignb[0];

    // ---- phase 1: softmax row (wave w -> row w of this n-tile) ----
    {
        const int   r    = wave;
        const float xr   = xw1[b * N_DIM + ntile * 16 + r];
        const float* __restrict__ aw2b  = aw2          + b * L_DIM;
        const float* __restrict__ smask = softmax_mask + b * L_DIM;
        float mx = -3.0e38f;
        for (int l = lane; l < L_DIM; l += 32) {
            float s = xr + aw2b[l] + ab;
            s = (s > 0.f) ? s : 0.01f * s;          // leaky relu
            s += smask[l];
            lds[r * ATT_STRIDE + l] = s;
            mx = fmaxf(mx, s);
        }
        for (int off = 16; off > 0; off >>= 1) mx = fmaxf(mx, __shfl_xor(mx, off, 32));
        float sum = 0.f;
        for (int l = lane; l < L_DIM; l += 32) {
            float e = __expf(lds[r * ATT_STRIDE + l] - mx);
            lds[r * ATT_STRIDE + l] = e;
            sum += e;
        }
        for (int off = 16; off > 0; off >>= 1) sum += __shfl_xor(sum, off, 32);
        const float inv = 1.f / sum;
        const float* __restrict__ amask = attend_mask + b * L_DIM;
        for (int l = lane; l < L_DIM; l += 32)
            lds[r * ATT_STRIDE + l] *= inv * amask[l];
    }
    __syncthreads();

    // ---- phase 2: ctx tile = attn(16x1024) @ a[b](1024x256), wave -> d-tile ----
    v8f c = {};
    {
        const int mA   = lane & 15;
        const int koff = (lane >> 4) << 1;
        const int dcol = wave * 16 + (lane & 15);
        const float* __restrict__ abp  = a + (size_t)b * L_DIM * D_DIM + dcol;
        const float* __restrict__ arow = lds + mA * ATT_STRIDE;
#pragma unroll 4
        for (int k = 0; k < L_DIM; k += 4) {
            v2f av, bv;
            av.x = arow[k + koff];
            av.y = arow[k + koff + 1];
            bv.x = abp[(size_t)(k + koff)     * D_DIM];
            bv.y = abp[(size_t)(k + koff + 1) * D_DIM];
            c = __builtin_amdgcn_wmma_f32_16x16x4_f32(false, av, false, bv,
                                                      (short)0, c, false, false);
        }
    }
    __syncthreads();

    // stage ctx tile in LDS (reuse attn storage)
    float* ctx = lds;  // 16 x 256
    {
        const int dcol = wave * 16 + (lane & 15);
        const int mb   = (lane >> 4) << 3;
#pragma unroll
        for (int r2 = 0; r2 < 8; ++r2)
            ctx[(mb + r2) * D_DIM + dcol] = c[r2];
    }
    __syncthreads();

    // ---- phase 3: layernorm, wave per row ----
    {
        const int r = wave;
        float s = 0.f, s2 = 0.f;
        for (int d = lane; d < D_DIM; d += 32) {
            float x = ctx[r * D_DIM + d];
            s += x; s2 += x * x;
        }
        for (int off = 16; off > 0; off >>= 1) {
            s  += __shfl_xor(s,  off, 32);
            s2 += __shfl_xor(s2, off, 32);
        }
        const float mu   = s * (1.f / D_DIM);
        const float var  = s2 * (1.f / D_DIM) - mu * mu;
        const float rstd = rsqrtf(var + 1e-5f);
        float* __restrict__ orow = out + (size_t)(b * N_DIM + ntile * 16 + r) * D_DIM;
        for (int d = lane; d < D_DIM; d += 32)
            orow[d] = (ctx[r * D_DIM + d] - mu) * rstd * gamma[d] + beta[d];
    }
}

// ---------------------------------------------------------------------------
extern "C" void kernel_launch(void* const* d_in, const int* in_sizes, int n_in,
                              void* d_out, int out_size, void* d_ws, size_t ws_size,
                              hipStream_t stream) {
    const float* mol    = (const float*)d_in[0];   // (16,256,256)
    const float* atom   = (const float*)d_in[1];   // (16,1024,256)
    const float* amask  = (const float*)d_in[2];   // (16,1024)
    const float* smask  = (const float*)d_in[3];   // (16,1024)
    const float* Wmol   = (const float*)d_in[4];   // (256,256)
    const float* bmol   = (const float*)d_in[5];   // (256,)
    const float* Wnb    = (const float*)d_in[6];   // (256,256)
    const float* bnb    = (const float*)d_in[7];   // (256,)
    const float* alignw = (const float*)d_in[8];   // (512,)
    const float* alignb = (const float*)d_in[9];   // scalar
    const float* gamma  = (const float*)d_in[10];  // (256,)
    const float* beta   = (const float*)d_in[11];  // (256,)
    float* out = (float*)d_out;

    // workspace layout
    float* ws  = (float*)d_ws;
    float* a   = ws;                                   // 16*1024*256 = 4194304 floats
    float* aw2 = a   + (size_t)B_DIM * L_DIM * D_DIM;  // 16384
    float* xw1 = aw2 + B_DIM * L_DIM;                  // 4096
    float* v   = xw1 + B_DIM * N_DIM;                  // 256
    float* u   = v   + D_DIM;                          // 256
    float* c12 = u   + D_DIM;                          // 2

    precompute_kernel<<<1, 256, 0, stream>>>(Wmol, Wnb, alignw, bmol, bnb, v, u, c12);
    proj_a_kernel<<<(B_DIM * L_DIM) / 16, 512, 0, stream>>>(atom, Wnb, bnb, a);
    dots_kernel<<<(B_DIM * N_DIM + B_DIM * L_DIM) / 256, 256, 0, stream>>>(mol, atom, v, u, c12, xw1, aw2);
    attn_kernel<<<B_DIM * 16, 512, 16 * ATT_STRIDE * sizeof(float), stream>>>(
        a, aw2, xw1, amask, smask, alignb, gamma, beta, out);
}